// Voxelization_87144886436561
// MI455X (gfx1250) — compile-verified
//
#include <hip/hip_runtime.h>
#include <math.h>

// Problem constants (from reference): features [16,32768,64] f32, coords [16,32768,3] f32
#define B_   16
#define N_   32768          // 2^15
#define C_   64             // 2^6
#define R_   32
#define V_   (R_ * R_ * R_) // 32768 = 2^15
#define EPS_ 1e-6f

typedef __attribute__((ext_vector_type(2))) float v2f;
typedef __attribute__((ext_vector_type(8))) float v8f;

// Wave(32)-wide sum. On gfx1250 use V_WMMA_F32_16X16X4_F32 with an all-ones A
// matrix: D[m][n] = sum_k B[k][n] for every row m, so every lane's acc[0] holds
// one of 16 column sums; a 4-step xor-shuffle over lanes yields the grand total
// = sum of all 64 B elements (32 lane values + 32 zeros). This is layout-
// invariant, so it is exact regardless of the B operand's VGPR mapping.
__device__ __forceinline__ float wave_sum(float v) {
#if defined(__HIP_DEVICE_COMPILE__) && __has_builtin(__builtin_amdgcn_wmma_f32_16x16x4_f32)
    v2f a;  a[0] = 1.0f; a[1] = 1.0f;       // all-ones A (16x4)
    v2f bm; bm[0] = v;   bm[1] = 0.0f;      // lane value + zero pad (4x16 B)
    v8f acc = {};
    acc = __builtin_amdgcn_wmma_f32_16x16x4_f32(
        /*neg_a=*/false, a, /*neg_b=*/false, bm,
        /*c_mod=*/(short)0, acc, /*reuse_a=*/false, /*reuse_b=*/false);
    float r = acc[0];                        // colsum_(lane&15) in every lane
#pragma unroll
    for (int m = 1; m < 16; m <<= 1) r += __shfl_xor(r, m, 32);
    return r;
#else
#pragma unroll
    for (int m = 1; m < 32; m <<= 1) v += __shfl_xor(v, m, 32);
    return v;
#endif
}

__global__ __launch_bounds__(256) void k_zero(float4* __restrict__ p, long long n4) {
    long long t = (long long)blockIdx.x * blockDim.x + threadIdx.x;
    if (t < n4) p[t] = make_float4(0.f, 0.f, 0.f, 0.f);
}

// One block (256 threads = 8 waves) per batch: mean of coords over N points.
__global__ __launch_bounds__(256) void k_coord_mean(const float* __restrict__ coords,
                                                    float* __restrict__ stats) {
    const int b = blockIdx.x;
    const float* cb = coords + (size_t)b * N_ * 3;
    float sx = 0.f, sy = 0.f, sz = 0.f;
    for (int i = threadIdx.x; i < N_; i += 256) {
        sx += cb[3 * i + 0];
        sy += cb[3 * i + 1];
        sz += cb[3 * i + 2];
    }
    sx = wave_sum(sx); sy = wave_sum(sy); sz = wave_sum(sz);
    __shared__ float red[8][3];
    const int w = threadIdx.x >> 5, l = threadIdx.x & 31;
    if (l == 0) { red[w][0] = sx; red[w][1] = sy; red[w][2] = sz; }
    __syncthreads();
    if (threadIdx.x == 0) {
        float tx = 0.f, ty = 0.f, tz = 0.f;
        for (int i = 0; i < 8; ++i) { tx += red[i][0]; ty += red[i][1]; tz += red[i][2]; }
        stats[b * 8 + 0] = tx * (1.0f / N_);
        stats[b * 8 + 1] = ty * (1.0f / N_);
        stats[b * 8 + 2] = tz * (1.0f / N_);
    }
}

// One block per batch: max ||coord - mean||^2, then store 1/(2*max_norm + eps).
__global__ __launch_bounds__(256) void k_coord_scale(const float* __restrict__ coords,
                                                     float* __restrict__ stats) {
    const int b = blockIdx.x;
    const float* cb = coords + (size_t)b * N_ * 3;
    const float mx = stats[b * 8 + 0], my = stats[b * 8 + 1], mz = stats[b * 8 + 2];
    float mn = 0.f;
    for (int i = threadIdx.x; i < N_; i += 256) {
        const float dx = cb[3 * i + 0] - mx;
        const float dy = cb[3 * i + 1] - my;
        const float dz = cb[3 * i + 2] - mz;
        mn = fmaxf(mn, dx * dx + dy * dy + dz * dz);
    }
#pragma unroll
    for (int m = 1; m < 32; m <<= 1) mn = fmaxf(mn, __shfl_xor(mn, m, 32));
    __shared__ float red[8];
    const int w = threadIdx.x >> 5, l = threadIdx.x & 31;
    if (l == 0) red[w] = mn;
    __syncthreads();
    if (threadIdx.x == 0) {
        float t = 0.f;
        for (int i = 0; i < 8; ++i) t = fmaxf(t, red[i]);
        stats[b * 8 + 3] = 1.0f / (2.0f * sqrtf(t) + EPS_);
    }
}

// 16 lanes per point, 4 channels each (float4). Scatter directly into the
// channel-major output [B,C,V] with global f32 atomics; whole 134 MB
// accumulator fits in the 192 MB L2, so atomics resolve at L2.
__global__ __launch_bounds__(256) void k_scatter(const float* __restrict__ feats,
                                                 const float* __restrict__ coords,
                                                 const float* __restrict__ stats,
                                                 float* __restrict__ out1,
                                                 float* __restrict__ out2,
                                                 float* __restrict__ counts) {
    const long long t = (long long)blockIdx.x * blockDim.x + threadIdx.x;
    const int       q = (int)(t & 15);   // channel quad 0..15
    const long long p = t >> 4;          // global point id, < B_*N_
    const int       b = (int)(p >> 15);  // N_ = 2^15

    const float* cp = coords + p * 3;
    const float* st = stats + (size_t)b * 8;
    const float inv = st[3];
    const float nx = (cp[0] - st[0]) * inv + 0.5f;
    const float ny = (cp[1] - st[1]) * inv + 0.5f;
    const float nz = (cp[2] - st[2]) * inv + 0.5f;

    int ix = (int)rintf(nx * (float)(R_ - 1));  // round-to-nearest-even == jnp.round
    int iy = (int)rintf(ny * (float)(R_ - 1));
    int iz = (int)rintf(nz * (float)(R_ - 1));
    ix = min(max(ix, 0), R_ - 1);
    iy = min(max(iy, 0), R_ - 1);
    iz = min(max(iz, 0), R_ - 1);
    const int idx = (ix << 10) + (iy << 5) + iz;

    if (q == 0) {
        out2[p * 3 + 0] = nx;
        out2[p * 3 + 1] = ny;
        out2[p * 3 + 2] = nz;
        atomicAdd(&counts[((long long)b << 15) + idx], 1.0f);
    }

    const float4 f = *(const float4*)(feats + (p << 6) + (q << 2));
    float* dst = out1 + (((size_t)b * C_ + (size_t)q * 4) * V_) + idx;
    atomicAdd(dst + 0 * V_, f.x);
    atomicAdd(dst + 1 * V_, f.y);
    atomicAdd(dst + 2 * V_, f.z);
    atomicAdd(dst + 3 * V_, f.w);
}

// In-place: out[b,c,v] /= max(count[b,v], 1). float4-vectorized, coalesced.
__global__ __launch_bounds__(256) void k_finalize(float4* __restrict__ out1,
                                                  const float* __restrict__ counts) {
    const long long t = (long long)blockIdx.x * blockDim.x + threadIdx.x; // < B*C*V/4
    const long long e = t << 2;
    const int v = (int)(e & (long long)(V_ - 1));
    const long long bc = e >> 15;           // V_ = 2^15
    const int b = (int)(bc >> 6);           // C_ = 2^6
    float4 s = out1[t];
    const float4 cn = *(const float4*)(counts + ((long long)b << 15) + v);
    s.x /= fmaxf(cn.x, 1.f);
    s.y /= fmaxf(cn.y, 1.f);
    s.z /= fmaxf(cn.z, 1.f);
    s.w /= fmaxf(cn.w, 1.f);
    out1[t] = s;
}

extern "C" void kernel_launch(void* const* d_in, const int* in_sizes, int n_in,
                              void* d_out, int out_size, void* d_ws, size_t ws_size,
                              hipStream_t stream) {
    const float* feats  = (const float*)d_in[0];  // [16,32768,64] f32
    const float* coords = (const float*)d_in[1];  // [16,32768,3]  f32

    float* out1 = (float*)d_out;                        // [B,C,V] = 33,554,432 f32
    float* out2 = out1 + (size_t)B_ * C_ * V_;          // [B,N,3] =  1,572,864 f32

    float* counts = (float*)d_ws;                       // B*V f32 (2 MB)
    float* stats  = counts + (size_t)B_ * V_;           // B*8 f32 (mean xyz, inv-scale)

    const long long n4_out1 = (long long)B_ * C_ * V_ / 4;  // 8,388,608
    const long long n4_cnt  = (long long)B_ * V_ / 4;       // 131,072

    k_zero<<<(unsigned)((n4_out1 + 255) / 256), 256, 0, stream>>>((float4*)out1, n4_out1);
    k_zero<<<(unsigned)((n4_cnt  + 255) / 256), 256, 0, stream>>>((float4*)counts, n4_cnt);
    k_coord_mean <<<B_, 256, 0, stream>>>(coords, stats);
    k_coord_scale<<<B_, 256, 0, stream>>>(coords, stats);

    const long long tsc = (long long)B_ * N_ * 16;          // 8,388,608 threads
    k_scatter<<<(unsigned)(tsc / 256), 256, 0, stream>>>(feats, coords, stats,
                                                         out1, out2, counts);
    k_finalize<<<(unsigned)(n4_out1 / 256), 256, 0, stream>>>((float4*)out1, counts);
}